// GCN_39161511805099
// MI455X (gfx1250) — compile-verified
//
#include <hip/hip_runtime.h>

// ---------------------------------------------------------------------------
// GCN on MI455X (gfx1250, wave32).
// GEMMs use V_WMMA_F32_16X16X4_F32 (full-precision f32 matrix core path),
// with a wave-uniform fast-path epilogue (no per-store EXEC juggling) and
// dual accumulator chains in the K=64 GEMM to break the WMMA RAW chain.
// Edge gather/scatter is the memory-bound hot loop: float4 gathers (L2
// resident, 192MB L2 >> 12.8MB feature array) + non-returning f32 atomics.
// ---------------------------------------------------------------------------

typedef __attribute__((ext_vector_type(2))) float v2f;
typedef __attribute__((ext_vector_type(8))) float v8f;

#define HID 64

__device__ __forceinline__ float eluf(float v) {
    return v > 0.0f ? v : (__expf(v) - 1.0f);
}

// ---- degree / norm ---------------------------------------------------------

__global__ void k_init_deg(float* __restrict__ deg, int n) {
    int i = blockIdx.x * blockDim.x + threadIdx.x;
    if (i < n) deg[i] = 1.0f;   // self-loop contribution
}

__global__ void k_deg_edges(const int* __restrict__ dst, float* __restrict__ deg, int e) {
    int i = blockIdx.x * blockDim.x + threadIdx.x;
    if (i < e) atomicAdd(&deg[dst[i]], 1.0f);
}

__global__ void k_rsqrt(float* __restrict__ deg, int n) {
    int i = blockIdx.x * blockDim.x + threadIdx.x;
    if (i < n) deg[i] = rsqrtf(deg[i]);   // deg >= 1 always
}

// ---- shared GEMM epilogue: store 16x16 f32 D tile --------------------------
// C/D layout: VGPR v: lanes 0-15 -> M=v,  N=lane; lanes 16-31 -> M=v+8.

__device__ __forceinline__ void store_tile(float* __restrict__ out, const v8f& c,
                                           int tm, int half, int bcol, int n) {
    float* op = out + (size_t)(tm * 16 + half * 8) * HID + bcol;
    if (tm * 16 + 16 <= n) {           // wave-uniform fast path: full tile
#pragma unroll
        for (int v = 0; v < 8; ++v) op[(size_t)v * HID] = c[v];
    } else {                           // tail tile: per-row guard
#pragma unroll
        for (int v = 0; v < 8; ++v) {
            if (tm * 16 + half * 8 + v < n) op[(size_t)v * HID] = c[v];
        }
    }
}

// ---- GEMM1: h = x[N,4] @ W1[4,64] via one WMMA f32 16x16x4 per tile --------
// A layout (32-bit A 16x4): lanes 0-15 -> M=lane, VGPR{0,1} = K{0,1};
//                           lanes 16-31 -> M=lane-16, VGPR{0,1} = K{2,3}.
// B layout (4x16):          lanes 0-15 -> N=lane,  K{0,1}; lanes16-31 K{2,3}.

__global__ void k_gemm_xw1(const float* __restrict__ x, const float* __restrict__ W1,
                           float* __restrict__ h, int n) {
    const int lane = threadIdx.x & 31;
    const int wave = (blockIdx.x * blockDim.x + threadIdx.x) >> 5;
    const int mtiles = (n + 15) >> 4;
    const int tm = wave >> 2;        // 4 N-tiles cover HID=64
    const int tn = wave & 3;
    if (tm >= mtiles) return;        // wave-uniform: EXEC stays all-ones below

    const int half = lane >> 4;      // 0: K=0,1   1: K=2,3
    const int l15  = lane & 15;

    int arow = tm * 16 + l15; if (arow >= n) arow = n - 1;
    const float* ap = x + (size_t)arow * 4 + half * 2;
    v2f a; a.x = ap[0]; a.y = ap[1];

    const int bcol = tn * 16 + l15;
    v2f b;
    b.x = W1[(half * 2 + 0) * HID + bcol];
    b.y = W1[(half * 2 + 1) * HID + bcol];

    v8f c = {};
    c = __builtin_amdgcn_wmma_f32_16x16x4_f32(false, a, false, b, (short)0, c, false, false);

    store_tile(h, c, tm, half, bcol, n);
}

// ---- GEMM2: h2 = elu(h)[N,64] @ W2[64,64] ---------------------------------
// 16 WMMAs over K, split across two accumulator chains to relax the
// WMMA->WMMA RAW hazard (5-cycle class for f32), merged at the end.

__global__ void k_gemm_hw2(const float* __restrict__ hin, const float* __restrict__ W2,
                           float* __restrict__ hout, int n) {
    const int lane = threadIdx.x & 31;
    const int wave = (blockIdx.x * blockDim.x + threadIdx.x) >> 5;
    const int mtiles = (n + 15) >> 4;
    const int tm = wave >> 2;
    const int tn = wave & 3;
    if (tm >= mtiles) return;        // wave-uniform

    const int half = lane >> 4;
    const int l15  = lane & 15;

    int arow = tm * 16 + l15; if (arow >= n) arow = n - 1;
    const float* ap = hin + (size_t)arow * HID;
    const int bcol = tn * 16 + l15;

    v8f c0 = {}, c1 = {};
#pragma unroll
    for (int k = 0; k < 16; k += 2) {
        // even K-step -> chain 0
        int kk = k * 4 + half * 2;
        v2f a, b;
        a.x = eluf(ap[kk + 0]);      // ELU fused into A-operand load
        a.y = eluf(ap[kk + 1]);
        b.x = W2[(kk + 0) * HID + bcol];
        b.y = W2[(kk + 1) * HID + bcol];
        c0 = __builtin_amdgcn_wmma_f32_16x16x4_f32(false, a, false, b, (short)0, c0, false, false);
        // odd K-step -> chain 1 (independent of chain 0)
        kk += 4;
        a.x = eluf(ap[kk + 0]);
        a.y = eluf(ap[kk + 1]);
        b.x = W2[(kk + 0) * HID + bcol];
        b.y = W2[(kk + 1) * HID + bcol];
        c1 = __builtin_amdgcn_wmma_f32_16x16x4_f32(false, a, false, b, (short)0, c1, false, false);
    }
    c0 = c0 + c1;

    store_tile(hout, c0, tm, half, bcol, n);
}

// ---- out[i] = dis[i]^2 * h[i] + bias  (self-loop term + bias) --------------

__global__ void k_init_out(const float* __restrict__ h, const float* __restrict__ dis,
                           const float* __restrict__ bias, float* __restrict__ out, int n) {
    int t = blockIdx.x * blockDim.x + threadIdx.x;   // n*16 threads, float4 each
    if (t >= n * 16) return;
    const int i  = t >> 4;
    const int j4 = (t & 15) << 2;
    const float w = dis[i] * dis[i];
    const float4 v = *(const float4*)(h + (size_t)i * HID + j4);
    float* op = out + (size_t)i * HID + j4;
    op[0] = v.x * w + bias[j4 + 0];
    op[1] = v.y * w + bias[j4 + 1];
    op[2] = v.z * w + bias[j4 + 2];
    op[3] = v.w * w + bias[j4 + 3];
}

// ---- hot loop: out[d] += dis[s]*dis[d] * h[s]  (16 lanes per edge) ---------

__global__ void k_scatter(const int* __restrict__ src, const int* __restrict__ dst,
                          const float* __restrict__ dis, const float* __restrict__ h,
                          float* __restrict__ out, int e) {
    int t = blockIdx.x * blockDim.x + threadIdx.x;   // e*16 threads
    if (t >= e * 16) return;
    const int ed = t >> 4;
    const int j4 = (t & 15) << 2;
    const int s = src[ed];
    const int d = dst[ed];
    const float nm = dis[s] * dis[d];
    const float4 v = *(const float4*)(h + (size_t)s * HID + j4);
    float* op = out + (size_t)d * HID + j4;
    atomicAdd(op + 0, v.x * nm);
    atomicAdd(op + 1, v.y * nm);
    atomicAdd(op + 2, v.z * nm);
    atomicAdd(op + 3, v.w * nm);
}

// ---- scatter-mean pooling (ELU fused) --------------------------------------

__global__ void k_pool(const float* __restrict__ h, const int* __restrict__ batch,
                       float* __restrict__ pool, float* __restrict__ cnt, int n) {
    int t = blockIdx.x * blockDim.x + threadIdx.x;   // n*16 threads
    if (t >= n * 16) return;
    const int i  = t >> 4;
    const int j4 = (t & 15) << 2;
    const int g = batch[i];
    const float4 v = *(const float4*)(h + (size_t)i * HID + j4);
    float* pp = pool + (size_t)g * HID + j4;
    atomicAdd(pp + 0, eluf(v.x));
    atomicAdd(pp + 1, eluf(v.y));
    atomicAdd(pp + 2, eluf(v.z));
    atomicAdd(pp + 3, eluf(v.w));
    if ((t & 15) == 0) atomicAdd(&cnt[g], 1.0f);
}

// ---- final: out[g] = (pool[g]/max(cnt,1)) . W3 + b3 ------------------------

__global__ void k_final(const float* __restrict__ pool, const float* __restrict__ cnt,
                        const float* __restrict__ W3, const float* __restrict__ b3,
                        float* __restrict__ out, int g) {
    int i = blockIdx.x * blockDim.x + threadIdx.x;
    if (i >= g) return;
    const float inv = 1.0f / fmaxf(cnt[i], 1.0f);
    float acc = b3[0];
#pragma unroll 8
    for (int j = 0; j < HID; ++j) acc += pool[(size_t)i * HID + j] * inv * W3[j];
    out[i] = acc;
}

// ---------------------------------------------------------------------------

extern "C" void kernel_launch(void* const* d_in, const int* in_sizes, int n_in,
                              void* d_out, int out_size, void* d_ws, size_t ws_size,
                              hipStream_t stream) {
    const float* x     = (const float*)d_in[0];
    const int*   ei    = (const int*)d_in[1];
    const int*   batch = (const int*)d_in[2];
    const float* W1    = (const float*)d_in[3];
    const float* b1    = (const float*)d_in[4];
    const float* W2    = (const float*)d_in[5];
    const float* b2    = (const float*)d_in[6];
    const float* W3    = (const float*)d_in[7];
    const float* b3    = (const float*)d_in[8];
    float* out = (float*)d_out;

    const int n = in_sizes[2];          // N_NODES
    const int e = in_sizes[1] / 2;      // N_EDGES
    const int g = out_size;             // N_GRAPHS
    const int* src = ei;
    const int* dst = ei + e;

    // workspace layout (floats): dis[n] | bufA[n*64] | bufB[n*64] | pool[g*64] | cnt[g]
    float* ws   = (float*)d_ws;
    float* dis  = ws;
    float* bufA = dis + n;
    float* bufB = bufA + (size_t)n * HID;
    float* pool = bufB + (size_t)n * HID;
    float* cnt  = pool + (size_t)g * HID;

    const int B = 256;
    const int mtiles  = (n + 15) / 16;
    const int gblocks = (mtiles * 4 + 3) / 4;   // 128 thr = 4 waves = 4 tiles/block

    // symmetric normalization
    k_init_deg <<<(n + B - 1) / B, B, 0, stream>>>(dis, n);
    k_deg_edges<<<(e + B - 1) / B, B, 0, stream>>>(dst, dis, e);
    k_rsqrt    <<<(n + B - 1) / B, B, 0, stream>>>(dis, n);

    // layer 1: h1 = x@W1 (WMMA); out1 = aggregate + b1
    k_gemm_xw1<<<gblocks, 128, 0, stream>>>(x, W1, bufA, n);
    k_init_out<<<(n * 16 + B - 1) / B, B, 0, stream>>>(bufA, dis, b1, bufB, n);
    k_scatter <<<(e * 16 + B - 1) / B, B, 0, stream>>>(src, dst, dis, bufA, bufB, e);

    // layer 2: h2 = elu(out1)@W2 (WMMA, ELU fused); out2 = aggregate + b2
    k_gemm_hw2<<<gblocks, 128, 0, stream>>>(bufB, W2, bufA, n);
    k_init_out<<<(n * 16 + B - 1) / B, B, 0, stream>>>(bufA, dis, b2, bufB, n);
    k_scatter <<<(e * 16 + B - 1) / B, B, 0, stream>>>(src, dst, dis, bufA, bufB, e);

    // pooling (re-zero accumulators every call: graph replays must be identical)
    hipMemsetAsync(pool, 0, (size_t)(g * HID + g) * sizeof(float), stream);
    k_pool <<<(n * 16 + B - 1) / B, B, 0, stream>>>(bufB, batch, pool, cnt, n);
    k_final<<<(g + B - 1) / B, B, 0, stream>>>(pool, cnt, W3, b3, out, g);
}